// OpenPosePredictor_12077448036547
// MI455X (gfx1250) — compile-verified
//
#include <hip/hip_runtime.h>

// ---------------- CDNA5 feature probes (device pass only) ----------------
#if defined(__HIP_DEVICE_COMPILE__) && __has_builtin(__builtin_amdgcn_wmma_f32_16x16x4_f32)
#define HAS_WMMA4 1
#else
#define HAS_WMMA4 0
#endif

#if defined(__HIP_DEVICE_COMPILE__) && \
    __has_builtin(__builtin_amdgcn_global_load_async_to_lds_b32) && \
    __has_builtin(__builtin_amdgcn_s_wait_asynccnt)
#define HAS_ASYNC 1
#else
#define HAS_ASYNC 0
#endif

typedef __attribute__((ext_vector_type(2))) float v2f;
typedef __attribute__((ext_vector_type(8))) float v8f;

namespace {
constexpr int C = 18, H = 1080, W = 1920;
constexpr int RAD = 12;                       // gaussian radius (25 taps)
constexpr int TILE_H = 32, TILE_W = 64;       // output tile per workgroup
constexpr int IN_H = TILE_H + 2 * RAD;        // 56 staged rows
constexpr int IN_W = TILE_W + 2 * RAD;        // 88 staged cols
constexpr int RAW_STRIDE = IN_W + 1;          // 89: break LDS bank patterns
constexpr int TS = 41;                        // Toeplitz row stride (16x40 + pad)
constexpr float THRE1 = 0.1f;
constexpr long long NPIX = (long long)C * H * W;  // 37,324,800

// normalized gaussian, sigma=3, radius=12 (exp(-x^2/18)/Z, Z=7.519669)
constexpr float G[25] = {
    4.46114e-05f, 1.60095e-04f, 5.14108e-04f, 1.47727e-03f, 3.79860e-03f,
    8.74088e-03f, 1.79975e-02f, 3.31600e-02f, 5.46716e-02f, 8.06593e-02f,
    1.06486e-01f, 1.25801e-01f, 1.32985e-01f, 1.25801e-01f, 1.06486e-01f,
    8.06593e-02f, 5.46716e-02f, 3.31600e-02f, 1.79975e-02f, 8.74088e-03f,
    3.79860e-03f, 1.47727e-03f, 5.14108e-04f, 1.60095e-04f, 4.46114e-05f};
}  // namespace

// -------------------------------------------------------------------------
// Kernel A: fused separable gaussian per 32x64 tile.
//   stage (async->LDS, symmetric-reflected) -> horizontal VALU pass ->
//   vertical pass as T(16x40) x Hconv(40x16) via 10x v_wmma_f32_16x16x4_f32
// -------------------------------------------------------------------------
__global__ __launch_bounds__(256) void conv_kernel(const float* __restrict__ heat,
                                                   float* __restrict__ enh) {
  __shared__ float s_raw[IN_H * RAW_STRIDE];  // 56*89*4  ~ 19.9 KB
  __shared__ float s_h[IN_H * TILE_W];        // 56*64*4  ~ 14.3 KB
  __shared__ float s_T[16 * TS];              // 16*41*4  ~  2.6 KB
  __shared__ float s_g[32];

  const int tid = threadIdx.x;
  const int x0 = blockIdx.x * TILE_W;
  const int y0 = blockIdx.y * TILE_H;
  const int ch = blockIdx.z;
  const float* src = heat + (size_t)ch * H * W;

  // ---- stage input tile with symmetric (edge-inclusive) reflection ----
  for (int i = tid; i < IN_H * IN_W; i += 256) {
    const int r = i / IN_W;
    const int c = i - r * IN_W;
    int gr = y0 + r - RAD;
    gr = (gr < 0) ? (-1 - gr) : gr;
    gr = (gr >= H) ? (2 * H - 1 - gr) : gr;
    int gc = x0 + c - RAD;
    gc = (gc < 0) ? (-1 - gc) : gc;
    gc = (gc >= W) ? (2 * W - 1 - gc) : gc;
#if HAS_ASYNC
    typedef __attribute__((address_space(1))) int* gas_t;
    typedef __attribute__((address_space(3))) int* las_t;
    __builtin_amdgcn_global_load_async_to_lds_b32(
        (gas_t)(src + (size_t)gr * W + gc),
        (las_t)&s_raw[r * RAW_STRIDE + c], /*offset=*/0, /*cpol=*/0);
#else
    s_raw[r * RAW_STRIDE + c] = src[(size_t)gr * W + gc];
#endif
  }
  if (tid < 25) s_g[tid] = G[tid];
#if HAS_ASYNC
  __builtin_amdgcn_s_wait_asynccnt(0);
#endif
  __syncthreads();

  // ---- horizontal pass: hconv[r][c], r in [0,56), c in [0,64) ----
  for (int i = tid; i < IN_H * TILE_W; i += 256) {
    const int r = i >> 6;
    const int c = i & 63;
    const float* row = &s_raw[r * RAW_STRIDE + c];
    float acc = 0.f;
#pragma unroll
    for (int t = 0; t < 25; ++t) acc = fmaf(G[t], row[t], acc);
    s_h[r * TILE_W + c] = acc;
  }
  // ---- build banded Toeplitz T[m][d] = g[d-m] for d-m in [0,24] ----
  for (int i = tid; i < 16 * 40; i += 256) {
    const int m = i / 40;
    const int d = i - m * 40;
    const int t = d - m;
    s_T[m * TS + d] = (t >= 0 && t <= 24) ? s_g[t] : 0.f;
  }
  __syncthreads();

  // ---- vertical pass: 8 waves, each one 16x16 output block ----
  const int lane = tid & 31;
  const int wv = tid >> 5;       // 0..7
  const int rb = wv >> 2;        // 0..1 : 16-row block
  const int cb = wv & 3;         // 0..3 : 16-col block
  const int m = lane & 15;       // M for A-chunk / N for B-chunk
  const int hi = lane >> 4;      // lane half selects K = {0,1} vs {2,3}

#if HAS_WMMA4
  v8f acc = {};
  const int base = rb * 16;      // hconv row offset of this block
#pragma unroll
  for (int j0 = 0; j0 < 40; j0 += 4) {
    const int k2 = j0 + 2 * hi;
    v2f a, b;
    a.x = s_T[m * TS + k2];                             // A[m, k2]
    a.y = s_T[m * TS + k2 + 1];                         // A[m, k2+1]
    b.x = s_h[(base + k2) * TILE_W + cb * 16 + m];      // B[k2,   n=m]
    b.y = s_h[(base + k2 + 1) * TILE_W + cb * 16 + m];  // B[k2+1, n=m]
    acc = __builtin_amdgcn_wmma_f32_16x16x4_f32(
        /*neg_a=*/false, a, /*neg_b=*/false, b,
        /*c_mod=*/(short)0, acc, /*reuse_a=*/false, /*reuse_b=*/false);
  }
  // D layout: VGPR v -> M = v + 8*hi, N = lane&15
  const int col = x0 + cb * 16 + m;
  const int row0 = y0 + rb * 16 + 8 * hi;   // first row this half-wave writes
  float* dst = enh + ((size_t)ch * H + row0) * W + col;
  if (y0 + TILE_H <= H) {
    // interior tile (33/34 of blocks): uniform, unguarded stores
#pragma unroll
    for (int v = 0; v < 8; ++v) {
      *dst = acc[v];
      dst += W;
    }
  } else {
    // bottom edge tile: per-row guard (divergent only here)
#pragma unroll
    for (int v = 0; v < 8; ++v) {
      if (row0 + v < H) dst[(size_t)v * W] = acc[v];
    }
  }
#else
  // VALU fallback (also taken by the host pass)
  for (int i = tid; i < TILE_H * TILE_W; i += 256) {
    const int rr = i >> 6;
    const int c = i & 63;
    const int row = y0 + rr;
    if (row >= H) continue;
    float acc = 0.f;
#pragma unroll
    for (int t = 0; t < 25; ++t) acc = fmaf(G[t], s_h[(rr + t) * TILE_W + c], acc);
    enh[((size_t)ch * H + row) * W + x0 + c] = acc;
  }
  (void)rb; (void)cb; (void)m; (void)hi; (void)lane; (void)wv;
#endif
}

// -------------------------------------------------------------------------
// Kernel B: 4-neighbor peak mask (OOB neighbor = 0) + masked scores.
// enh re-read mostly hits the 192MB L2 (whole map is 149MB).
// -------------------------------------------------------------------------
__global__ __launch_bounds__(128) void peak_kernel(const float* __restrict__ heat,
                                                   const float* __restrict__ enh,
                                                   float* __restrict__ peaks,
                                                   float* __restrict__ pvals) {
  const int c = blockIdx.x * 128 + threadIdx.x;
  const int r = blockIdx.y;
  const int ch = blockIdx.z;
  const size_t idx = ((size_t)ch * H + r) * W + c;
  const float x = enh[idx];
  const float lft = (c > 0) ? enh[idx - 1] : 0.f;
  const float rgt = (c + 1 < W) ? enh[idx + 1] : 0.f;
  const float up = (r > 0) ? enh[idx - W] : 0.f;
  const float dn = (r + 1 < H) ? enh[idx + W] : 0.f;
  const bool p = (x >= lft) && (x >= rgt) && (x >= up) && (x >= dn) && (x > THRE1);
  peaks[idx] = p ? 1.f : 0.f;
  pvals[idx] = p ? heat[idx] : 0.f;
}

// -------------------------------------------------------------------------
extern "C" void kernel_launch(void* const* d_in, const int* in_sizes, int n_in,
                              void* d_out, int out_size, void* d_ws, size_t ws_size,
                              hipStream_t stream) {
  (void)in_sizes; (void)n_in; (void)out_size; (void)d_ws; (void)ws_size;
  const float* heat = (const float*)d_in[0];
  float* out = (float*)d_out;
  float* enh = out;                 // output 0: heatmap_enh
  float* peaks = out + NPIX;        // output 1: peaks_binary (as 0/1 float)
  float* pvals = out + 2 * NPIX;    // output 2: peak_values

  dim3 gA(W / TILE_W, (H + TILE_H - 1) / TILE_H, C);  // 30 x 34 x 18
  conv_kernel<<<gA, 256, 0, stream>>>(heat, enh);

  dim3 gB(W / 128, H, C);                             // 15 x 1080 x 18
  peak_kernel<<<gB, 128, 0, stream>>>(heat, enh, peaks, pvals);
}